// TokenEncoder_9105330668034
// MI455X (gfx1250) — compile-verified
//
#include <hip/hip_runtime.h>

// Problem constants (match reference)
#define B_BATCH 128
#define L_TOK   64
#define D_DIM   10000
#define JTILES  (D_DIM / 16)          // 625 column tiles of 16
#define WAVES_PER_BLOCK 8
#define THREADS (WAVES_PER_BLOCK * 32)

typedef __attribute__((ext_vector_type(2))) float v2f;
typedef __attribute__((ext_vector_type(8))) float v8f;

// out[b, j] = sign( sum_l item_mem[tok[b,l], (j - l) mod D] )
// Reduction over l done on the matrix pipe: V_WMMA_F32_16X16X4_F32 with
// A == all-ones  =>  D[m,n] = sum_k B[k,n] + C[m,n]   (pure f32 column-sum).
// One wave owns one (batch, 16-column) tile; 16 chained WMMAs cover l=0..63.
__global__ __launch_bounds__(THREADS)
void hdc_encode_wmma(const int* __restrict__ tokens,
                     const float* __restrict__ item_mem,
                     int* __restrict__ out) {
  const int lane = threadIdx.x & 31;
  const int wave = threadIdx.x >> 5;

  int w = blockIdx.x * WAVES_PER_BLOCK + wave;
  w = __builtin_amdgcn_readfirstlane(w);      // force scalar b / j0
  const int b  = w / JTILES;                  // batch row
  const int j0 = (w - b * JTILES) * 16;       // first output column of tile

  const int col  = j0 + (lane & 15);          // this lane's output column
  const int krow = (lane >> 4) << 1;          // lanes 0-15 -> K rows {0,1}; 16-31 -> {2,3}
  const int* __restrict__ tok = tokens + b * L_TOK;

  v2f a; a.x = 1.0f; a.y = 1.0f;              // all-ones A matrix (16x4)
  v8f c = {};                                 // f32 accumulator (16x16)

  #pragma unroll 4
  for (int l0 = 0; l0 < L_TOK; l0 += 4) {
    // 4 tokens for this K-chunk: uniform across the wave -> scalar loads
    const int t0 = tok[l0 + 0];
    const int t1 = tok[l0 + 1];
    const int t2 = tok[l0 + 2];
    const int t3 = tok[l0 + 3];
    const int tA = krow ? t2 : t0;
    const int tB = krow ? t3 : t1;

    const int lA = l0 + krow;
    const int lB = lA + 1;
    int cA = col - lA; if (cA < 0) cA += D_DIM;   // circular roll, no int mod
    int cB = col - lB; if (cB < 0) cB += D_DIM;

    v2f bv;                                       // B operand: 2 l-rows / lane
    bv.x = item_mem[tA * D_DIM + cA];
    bv.y = item_mem[tB * D_DIM + cB];

    // D = A(ones) x B + C  ==> per-column partial sums of 4 l-slices
    c = __builtin_amdgcn_wmma_f32_16x16x4_f32(
        /*neg_a=*/false, a, /*neg_b=*/false, bv,
        /*c_mod=*/(short)0, c, /*reuse_a=*/false, /*reuse_b=*/false);
  }

  // C VGPR0, lanes 0-15 hold row M=0 == the full column sums for this tile.
  if (lane < 16) {
    out[b * D_DIM + col] = (c[0] > 0.0f) ? 1 : -1;
  }
}

extern "C" void kernel_launch(void* const* d_in, const int* in_sizes, int n_in,
                              void* d_out, int out_size, void* d_ws, size_t ws_size,
                              hipStream_t stream) {
  const int*   tokens   = (const int*)d_in[0];    // [128, 64] int32
  const float* item_mem = (const float*)d_in[1];  // [1024, 10000] f32
  int*         out      = (int*)d_out;            // [128, 10000] int32 (+/-1)

  const int total_waves = B_BATCH * JTILES;              // 80000
  const int blocks = total_waves / WAVES_PER_BLOCK;      // 10000 (exact)
  hdc_encode_wmma<<<blocks, THREADS, 0, stream>>>(tokens, item_mem, out);
}